// CrystalGraphConvNet_16071767621932
// MI455X (gfx1250) — compile-verified
//
#include <hip/hip_runtime.h>
#include <hip/hip_bf16.h>
#include <math.h>

typedef float v2f __attribute__((ext_vector_type(2)));
typedef float v8f __attribute__((ext_vector_type(8)));

#define N_ATOMSc 512
#define Mn 132
#define ORIG_Fc 92
#define ATOM_Fc 106
#define NBR_Fc 41
#define TWOF 212          // 2*ATOM_F
#define KCAT 253          // 2*ATOM_F + NBR_F
#define KPAD 256
#define H_FEAc 128
#define N0c 16
#define ROWS_NM (N_ATOMSc*Mn)      // 67584
#define PAIR_ROWS (N0c*Mn*Mn)      // 278784
#define KPc 147                    // ATOM_F + NBR_F
#define KPP 148
#define EPSc 1e-5f
#define MIX_ELEMS (N0c*Mn*5*Mn*5)  // 6969600

// LDS strides chosen so the two half-waves (K offset +2 rows) land on
// disjoint 16-bank groups: A stride 40 (2*40=80 = +16 mod 64),
// B stride 20 (2*20=40 = +40 mod 64).
#define ASTR 40
#define BSTR 20

__device__ __forceinline__ float softplusf(float x){ return x > 20.f ? x : log1pf(expf(x)); }
__device__ __forceinline__ float sigmoidf(float x){ return 1.f/(1.f+expf(-x)); }

// ---------------- embed: h = atom_fea @ We + be ----------------
__global__ void __launch_bounds__(128) embed_kernel(
    const float* __restrict__ atom_fea, const float* __restrict__ We,
    const float* __restrict__ be, float* __restrict__ h)
{
  __shared__ float row[ORIG_Fc];
  int i = blockIdx.x;
  for(int k=threadIdx.x;k<ORIG_Fc;k+=128) row[k]=atom_fea[i*ORIG_Fc+k];
  __syncthreads();
  int f=threadIdx.x;
  if(f<ATOM_Fc){
    float acc=be[f];
    for(int k=0;k<ORIG_Fc;k++) acc += row[k]*We[k*ATOM_Fc+f];
    h[i*ATOM_Fc+f]=acc;
  }
}

// ---------------- mask count (once) ----------------
__global__ void __launch_bounds__(256) count_kernel(const int* __restrict__ pad, float* __restrict__ cnt){
  __shared__ float s[256];
  float a=0.f;
  for(int r=threadIdx.x;r<ROWS_NM;r+=256) a += (float)pad[r];
  s[threadIdx.x]=a; __syncthreads();
  for(int st=128;st>0;st>>=1){ if(threadIdx.x<st) s[threadIdx.x]+=s[threadIdx.x+st]; __syncthreads(); }
  if(threadIdx.x==0) *cnt=s[0];
}

// ---------------- conv GEMM: gated = concat(h[i], h[nbr], nbr_fea) @ W + b ----------------
// 64 threads (2 waves), 32 rows/block, fp32 WMMA 16x16x4.
__global__ void __launch_bounds__(64) conv_gemm_kernel(
    const float* __restrict__ h, const float* __restrict__ nbr_fea,
    const int* __restrict__ nbr_idx, const float* __restrict__ W,
    const float* __restrict__ bias, float* __restrict__ gated)
{
  __shared__ float As[KPAD*ASTR];   // K-major: As[k*ASTR + r]   (41 KB)
  __shared__ float Bs[KPAD*BSTR];   // Bs[k*BSTR + n]            (20.5 KB)
  __shared__ int rI[32], rN[32];
  int tid = threadIdx.x;
  int rb = blockIdx.x*32;
  if(tid<32){ int g = rb+tid; rI[tid]=g/Mn; rN[tid]=nbr_idx[g]; }
  __syncthreads();
  // row-consecutive fill: per wave-iteration k is uniform, lanes write
  // consecutive r -> conflict-free LDS stores.
  for(int idx=tid; idx<32*KPAD; idx+=64){
    int k = idx>>5, r = idx&31;
    int g = rb+r;
    float v=0.f;
    if(k<ATOM_Fc)        v = h[rI[r]*ATOM_Fc + k];
    else if(k<2*ATOM_Fc) v = h[rN[r]*ATOM_Fc + (k-ATOM_Fc)];
    else if(k<KCAT)      v = nbr_fea[(size_t)g*NBR_Fc + (k-2*ATOM_Fc)];
    As[k*ASTR + r] = v;
  }
  int lane = tid & 31, w = tid >> 5;
  int nl = lane & 15, hi = lane >> 4;
  int rr = nl + w*16;
  for(int ct=0; ct<14; ct++){
    __syncthreads();
    for(int idx=tid; idx<KPAD*16; idx+=64){
      int k = idx>>4, n = idx&15;
      int col = ct*16+n;
      Bs[k*BSTR+n] = (k<KCAT && col<TWOF) ? W[k*TWOF+col] : 0.f;
    }
    __syncthreads();
    v8f c;
    #pragma unroll
    for(int e=0;e<8;e++) c[e]=0.f;
    for(int k0=0;k0<KPAD;k0+=4){
      int kb = k0 + hi*2;
      v2f a, b;
      a.x = As[kb*ASTR+rr];     a.y = As[(kb+1)*ASTR+rr];
      b.x = Bs[kb*BSTR+nl];     b.y = Bs[(kb+1)*BSTR+nl];
      c = __builtin_amdgcn_wmma_f32_16x16x4_f32(false, a, false, b, (short)0, c, false, false);
    }
    int col = ct*16+nl;
    if(col<TWOF){
      float bv = bias[col];
      #pragma unroll
      for(int v=0;v<8;v++){
        int grow = rb + w*16 + v + 8*hi;
        gated[(size_t)grow*TWOF + col] = c[v] + bv;
      }
    }
  }
}

// ---------------- masked BN column stats (atomic accumulate) ----------------
__global__ void __launch_bounds__(256) stats_kernel(
    const float* __restrict__ gated, const int* __restrict__ pad, float* __restrict__ stats)
{
  __shared__ float ps[2*TWOF];
  __shared__ float msk[64];
  int tid=threadIdx.x;
  for(int c=tid;c<2*TWOF;c+=256) ps[c]=0.f;
  int rb = blockIdx.x*64;
  if(tid<64) msk[tid] = (float)pad[rb+tid];
  __syncthreads();
  for(int idx=tid; idx<64*TWOF; idx+=256){
    int r = idx/TWOF, c = idx - r*TWOF;
    if(msk[r]!=0.f){
      float v = gated[(size_t)(rb+r)*TWOF+c];
      atomicAdd(&ps[c], v);
      atomicAdd(&ps[TWOF+c], v*v);
    }
  }
  __syncthreads();
  for(int c=tid;c<2*TWOF;c+=256) if(ps[c]!=0.f) atomicAdd(&stats[c], ps[c]);
}

// ---------------- BN-apply + sigmoid*softplus gated neighbor sum ----------------
__global__ void __launch_bounds__(128) gate_sum_kernel(
    const float* __restrict__ gated, const int* __restrict__ pad,
    const float* __restrict__ stats, const float* __restrict__ cntp,
    const float* __restrict__ g1, const float* __restrict__ b1,
    float* __restrict__ summed)
{
  __shared__ float scale[TWOF], shift[TWOF];
  int tid=threadIdx.x;
  float cnt = *cntp;
  for(int c=tid;c<TWOF;c+=128){
    float mean = stats[c]/cnt;
    float var  = stats[TWOF+c]/cnt - mean*mean;
    float sc   = rsqrtf(var+EPSc)*g1[c];
    scale[c]=sc; shift[c]=b1[c]-mean*sc;
  }
  __syncthreads();
  int i = blockIdx.x;
  if(tid<ATOM_Fc){
    int f=tid;
    float acc=0.f;
    for(int m=0;m<Mn;m++){
      if(pad[i*Mn+m]){
        const float* row = &gated[(size_t)(i*Mn+m)*TWOF];
        float nf = row[f]*scale[f]+shift[f];
        float nc = row[ATOM_Fc+f]*scale[ATOM_Fc+f]+shift[ATOM_Fc+f];
        acc += sigmoidf(nf)*softplusf(nc);
      }
    }
    summed[i*ATOM_Fc+f]=acc;
  }
}

// ---------------- plain BN over atoms + softplus residual ----------------
__global__ void __launch_bounds__(256) bn2_kernel(
    const float* __restrict__ summed, const float* __restrict__ hin,
    const float* __restrict__ g2, const float* __restrict__ b2,
    float* __restrict__ hout)
{
  __shared__ float s[256], q[256];
  int c = blockIdx.x, tid=threadIdx.x;
  float a0 = summed[tid*ATOM_Fc+c];
  float a1 = summed[(tid+256)*ATOM_Fc+c];
  s[tid]=a0+a1; q[tid]=a0*a0+a1*a1;
  __syncthreads();
  for(int st=128;st>0;st>>=1){ if(tid<st){ s[tid]+=s[tid+st]; q[tid]+=q[tid+st]; } __syncthreads(); }
  float mean = s[0]*(1.f/512.f);
  float var  = q[0]*(1.f/512.f) - mean*mean;
  float sc = rsqrtf(var+EPSc)*g2[c];
  float sh = b2[c]-mean*sc;
  hout[tid*ATOM_Fc+c]       = softplusf(hin[tid*ATOM_Fc+c]       + a0*sc + sh);
  hout[(tid+256)*ATOM_Fc+c] = softplusf(hin[(tid+256)*ATOM_Fc+c] + a1*sc + sh);
}

// ---------------- site neighbor gather ----------------
__global__ void __launch_bounds__(128) site_gather_kernel(
    const float* __restrict__ h, const int* __restrict__ cai,
    const int* __restrict__ site_idx, const int* __restrict__ nbr_idx,
    float* __restrict__ site_nbr)
{
  int bm = blockIdx.x;
  int b = bm/Mn;
  int sa = cai[b*32 + site_idx[b]];
  int a = nbr_idx[sa*Mn + (bm - b*Mn)];
  for(int f=threadIdx.x; f<ATOM_Fc; f+=128)
    site_nbr[(size_t)bm*ATOM_Fc+f] = h[(size_t)a*ATOM_Fc+f];
}

// ---------------- pair GEMM (147->128 WMMA) fused with fc_out (128->3) ----------------
// 2-column-tile register blocking: each A fragment feeds 2 WMMAs.
__global__ void __launch_bounds__(64) pair_gemm_kernel(
    const float* __restrict__ site_nbr, const float* __restrict__ tab_hd,
    const int* __restrict__ batch_cif, const float* __restrict__ Wc,
    const float* __restrict__ bc, const float* __restrict__ Wo,
    const float* __restrict__ bo, float* __restrict__ out3)
{
  __shared__ float As[KPP*ASTR];   // 23.7 KB; reused as sp[32][128] after GEMM
  __shared__ float Bs0[KPP*BSTR];  // 11.8 KB
  __shared__ float Bs1[KPP*BSTR];  // 11.8 KB
  __shared__ int rB[32], rBB[32], rIv[32], rJv[32];
  int tid=threadIdx.x;
  int rb = blockIdx.x*32;
  if(tid<32){
    int g=rb+tid;
    int b = g/(Mn*Mn); int rem = g - b*Mn*Mn;
    int i = rem/Mn;    int j = rem - i*Mn;
    rB[tid]=b; rBB[tid]=batch_cif[b]; rIv[tid]=i; rJv[tid]=j;
  }
  __syncthreads();
  for(int idx=tid; idx<32*KPP; idx+=64){
    int k = idx>>5, r = idx&31;
    float v;
    if(k<ATOM_Fc){
      float s0 = site_nbr[((size_t)rB[r]*Mn+rIv[r])*ATOM_Fc+k];
      float s1 = site_nbr[((size_t)rB[r]*Mn+rJv[r])*ATOM_Fc+k];
      v = softplusf(0.5f*(s0+s1));
    } else if(k<KPc){
      v = softplusf(tab_hd[(((size_t)rBB[r]*Mn+rIv[r])*Mn+rJv[r])*NBR_Fc + (k-ATOM_Fc)]);
    } else v = 0.f;
    As[k*ASTR+r] = v;
  }
  int lane=tid&31, w=tid>>5, nl=lane&15, hi=lane>>4;
  int rr = nl + w*16;
  v8f cf[8];
  for(int ctp=0; ctp<4; ctp++){
    __syncthreads();
    for(int idx=tid; idx<KPP*16; idx+=64){
      int k=idx>>4, n=idx&15;
      int col0 = ctp*32+n;
      float w0 = (k<KPc) ? Wc[k*H_FEAc + col0]      : 0.f;
      float w1 = (k<KPc) ? Wc[k*H_FEAc + col0 + 16] : 0.f;
      Bs0[k*BSTR+n] = w0;
      Bs1[k*BSTR+n] = w1;
    }
    __syncthreads();
    v8f c0, c1;
    #pragma unroll
    for(int e=0;e<8;e++){ c0[e]=0.f; c1[e]=0.f; }
    for(int k0=0;k0<KPP;k0+=4){
      int kb=k0+hi*2;
      v2f a,b0,b1;
      a.x =As[kb*ASTR+rr];     a.y =As[(kb+1)*ASTR+rr];
      b0.x=Bs0[kb*BSTR+nl];    b0.y=Bs0[(kb+1)*BSTR+nl];
      b1.x=Bs1[kb*BSTR+nl];    b1.y=Bs1[(kb+1)*BSTR+nl];
      c0 = __builtin_amdgcn_wmma_f32_16x16x4_f32(false, a, false, b0, (short)0, c0, false, false);
      c1 = __builtin_amdgcn_wmma_f32_16x16x4_f32(false, a, false, b1, (short)0, c1, false, false);
    }
    cf[2*ctp+0]=c0; cf[2*ctp+1]=c1;
  }
  __syncthreads();
  float* sp = As;   // reuse: 32*128 = 4096 <= KPP*ASTR
  #pragma unroll
  for(int ct=0;ct<8;ct++){
    int col = ct*16+nl;
    float bv = bc[col];
    #pragma unroll
    for(int v=0;v<8;v++){
      int rloc = w*16 + v + 8*hi;
      sp[rloc*H_FEAc + col] = softplusf(cf[ct][v] + bv);
    }
  }
  __syncthreads();
  for(int idx=tid; idx<96; idx+=64){
    int r = idx/3, o = idx - r*3;
    float acc = bo[o];
    for(int f=0;f<H_FEAc;f++) acc += sp[r*H_FEAc+f]*Wo[f*3+o];
    out3[(size_t)(rb+r)*3 + o] = acc;
  }
}

// ---------------- per-(b,i,j) mixing coefficients ----------------
__global__ void __launch_bounds__(256) coef_kernel(const float* __restrict__ out3, float* __restrict__ coef){
  int g = blockIdx.x*256+threadIdx.x;
  if(g<PAIR_ROWS){
    float o0=out3[(size_t)g*3+0], o1=out3[(size_t)g*3+1];
    float sp=softplusf(out3[(size_t)g*3+2]);
    coef[(size_t)g*5+0]=powf(sp, 2.f/3.5f);
    coef[(size_t)g*5+1]=sp;
    coef[(size_t)g*5+2]=powf(sp, 5.f/3.5f);
    coef[(size_t)g*5+3]=o0;
    coef[(size_t)g*5+4]=o1;
  }
}

// ---------------- streaming mix against 6 tables ----------------
__global__ void __launch_bounds__(256) mixed_kernel(
  const float* __restrict__ coef, const int* __restrict__ batch_cif,
  const float* __restrict__ hop, const float* __restrict__ pss,
  const float* __restrict__ pds, const float* __restrict__ pdd,
  const float* __restrict__ pgds, const float* __restrict__ pgdd,
  float* __restrict__ outp)
{
  int id = blockIdx.x*256+threadIdx.x;
  if(id < MIX_ELEMS){
    int t=id;
    int e2=t%5; t/=5; int j=t%Mn; t/=Mn; int e1=t%5; t/=5; int i=t%Mn; int b=t/Mn;
    int bb = batch_cif[b];
    size_t tix = ((((size_t)bb*Mn+i)*5+e1)*Mn+j)*5+e2;
    size_t bij = ((size_t)b*Mn+i)*Mn+j;
    const float* cfp = &coef[bij*5];
    float mx = cfp[0]*pss[tix] + cfp[1]*pds[tix] + cfp[2]*pdd[tix]
             + cfp[3]*pgds[tix] + cfp[4]*pgdd[tix];
    outp[id] = hop[tix]*mx;
  }
}

extern "C" void kernel_launch(void* const* d_in, const int* in_sizes, int n_in,
                              void* d_out, int out_size, void* d_ws, size_t ws_size,
                              hipStream_t stream)
{
  const float* atom_fea = (const float*)d_in[0];
  const float* nbr_fea  = (const float*)d_in[1];
  const int*   nbr_idx  = (const int*)d_in[2];
  const int*   pad      = (const int*)d_in[3];
  const int*   cai      = (const int*)d_in[4];
  const int*   site_idx = (const int*)d_in[5];
  const int*   bcif     = (const int*)d_in[6];
  const float* tab_hd   = (const float*)d_in[7];
  const float* hop      = (const float*)d_in[8];
  const float* pss      = (const float*)d_in[9];
  const float* pds      = (const float*)d_in[10];
  const float* pdd      = (const float*)d_in[11];
  const float* pgds     = (const float*)d_in[12];
  const float* pgdd     = (const float*)d_in[13];

  const float *eW,*eb,*cW,*cb,*oW,*ob;
  const float *fW[3],*fb[3],*g1[3],*b1[3],*g2[3],*b2[3];
  if(in_sizes[14] == 147*128){
    // jax pytree order (sorted dict keys): conv_to_fc, convs[l]{b1,b2,fc.W,fc.b,g1,g2}, embed, fc_out
    cW=(const float*)d_in[14]; cb=(const float*)d_in[15];
    for(int l=0;l<3;l++){
      int base=16+6*l;
      b1[l]=(const float*)d_in[base+0];
      b2[l]=(const float*)d_in[base+1];
      fW[l]=(const float*)d_in[base+2];
      fb[l]=(const float*)d_in[base+3];
      g1[l]=(const float*)d_in[base+4];
      g2[l]=(const float*)d_in[base+5];
    }
    eW=(const float*)d_in[34]; eb=(const float*)d_in[35];
    oW=(const float*)d_in[36]; ob=(const float*)d_in[37];
  } else {
    // insertion order: embed, convs[l]{fc.W,fc.b,g1,b1,g2,b2}, conv_to_fc, fc_out
    eW=(const float*)d_in[14]; eb=(const float*)d_in[15];
    for(int l=0;l<3;l++){
      int base=16+6*l;
      fW[l]=(const float*)d_in[base+0];
      fb[l]=(const float*)d_in[base+1];
      g1[l]=(const float*)d_in[base+2];
      b1[l]=(const float*)d_in[base+3];
      g2[l]=(const float*)d_in[base+4];
      b2[l]=(const float*)d_in[base+5];
    }
    cW=(const float*)d_in[34]; cb=(const float*)d_in[35];
    oW=(const float*)d_in[36]; ob=(const float*)d_in[37];
  }

  float* ws = (float*)d_ws;
  float* hA       = ws;
  float* hB       = hA + N_ATOMSc*ATOM_Fc;
  float* gated    = hB + N_ATOMSc*ATOM_Fc;
  float* stats    = gated + (size_t)ROWS_NM*TWOF;
  float* cnt      = stats + 2*TWOF;
  float* summed   = cnt + 8;
  float* site_nbr = summed + N_ATOMSc*ATOM_Fc;
  float* coef     = site_nbr + (size_t)N0c*Mn*ATOM_Fc;

  float* out_mixed = (float*)d_out;
  float* out3 = out_mixed + (size_t)MIX_ELEMS;

  embed_kernel<<<N_ATOMSc,128,0,stream>>>(atom_fea, eW, eb, hA);
  count_kernel<<<1,256,0,stream>>>(pad, cnt);
  float* hin=hA; float* hout=hB;
  for(int l=0;l<3;l++){
    hipMemsetAsync(stats, 0, 2*TWOF*sizeof(float), stream);
    conv_gemm_kernel<<<ROWS_NM/32,64,0,stream>>>(hin, nbr_fea, nbr_idx, fW[l], fb[l], gated);
    stats_kernel<<<ROWS_NM/64,256,0,stream>>>(gated, pad, stats);
    gate_sum_kernel<<<N_ATOMSc,128,0,stream>>>(gated, pad, stats, cnt, g1[l], b1[l], summed);
    bn2_kernel<<<ATOM_Fc,256,0,stream>>>(summed, hin, g2[l], b2[l], hout);
    float* tswap=hin; hin=hout; hout=tswap;
  }
  site_gather_kernel<<<N0c*Mn,128,0,stream>>>(hin, cai, site_idx, nbr_idx, site_nbr);
  pair_gemm_kernel<<<PAIR_ROWS/32,64,0,stream>>>(site_nbr, tab_hd, bcif, cW, cb, oW, ob, out3);
  coef_kernel<<<(PAIR_ROWS+255)/256,256,0,stream>>>(out3, coef);
  mixed_kernel<<<MIX_ELEMS/256,256,0,stream>>>(coef, bcif, hop, pss, pds, pdd, pgds, pgdd, out_mixed);
}